// GridEncoder_11209864642768
// MI455X (gfx1250) — compile-verified
//
#include <hip/hip_runtime.h>
#include <stdint.h>

typedef float    vf4  __attribute__((ext_vector_type(4)));
typedef float    vf2  __attribute__((ext_vector_type(2)));
typedef uint32_t v4u  __attribute__((ext_vector_type(4)));
typedef int      v8i  __attribute__((ext_vector_type(8)));
typedef int      v4i  __attribute__((ext_vector_type(4)));

static constexpr int      kL  = 16;
static constexpr int      kBlock = 256;
static constexpr uint32_t kP1 = 2654435761u;
static constexpr uint32_t kP2 = 805459861u;

// Static per-level constants mirroring build_offsets(): all levels hash,
// all hash sizes are powers of two.
__device__ constexpr uint32_t kOffset[kL] = {
    0u, 4096u, 36864u, 299008u, 823296u, 1347584u, 1871872u, 2396160u,
    2920448u, 3444736u, 3969024u, 4493312u, 5017600u, 5541888u, 6066176u, 6590464u};
__device__ constexpr uint32_t kMask[kL] = {
    4095u, 32767u, 262143u, 524287u, 524287u, 524287u, 524287u, 524287u,
    524287u, 524287u, 524287u, 524287u, 524287u, 524287u, 524287u, 524287u};

__global__ __launch_bounds__(kBlock)
void grid_encode_kernel(const float* __restrict__ in,
                        const vf2* __restrict__ emb,
                        float* __restrict__ out, int B)
{
    // Level-0 table: 4096 float2 = 32 KB, staged in LDS.
    __shared__ vf2 s_l0[4096];

#if defined(__gfx1250__) && __has_builtin(__builtin_amdgcn_tensor_load_to_lds) && \
    __has_builtin(__builtin_amdgcn_s_wait_tensorcnt)
    // TDM: DMA 32 KB (4096 x 8B elements, 1 row) from emb[0] into LDS.
    if (threadIdx.x < 32) {   // one wave issues; TDM ignores EXEC
        uint64_t ga  = (uint64_t)(uintptr_t)emb;
        uint32_t lds = (uint32_t)(uintptr_t)&s_l0[0];   // low 32 bits = LDS offset
        v4u g0;
        g0[0] = 1u;                                        // count=1, no gather mode
        g0[1] = lds;                                       // lds_addr (bytes)
        g0[2] = (uint32_t)ga;                              // global_addr[31:0]
        g0[3] = (uint32_t)((ga >> 32) & 0x01FFFFFFu)       // global_addr[56:32]
              | (2u << 30);                                // type = 2 ("image")
        v8i g1;
        g1[0] = (int)(3u << 16);                           // wg_mask=0, data_size=8B
        g1[1] = (int)(4096u << 16);                        // tensor_dim0[15:0]=4096
        g1[2] = (int)(1u << 16);                           // tensor_dim0 hi=0, tensor_dim1=1
        g1[3] = (int)(4096u << 16);                        // tile_dim0=4096
        g1[4] = 1;                                         // tile_dim1=1, tile_dim2=0
        g1[5] = 4096;                                      // tensor_dim0_stride lo
        g1[6] = 0;
        g1[7] = 0;
        v4i gz4 = {0, 0, 0, 0};                            // groups 2/3 unused (<=2D)
        v8i gz8 = {0, 0, 0, 0, 0, 0, 0, 0};                // extra group (clang-23 form)
        __builtin_amdgcn_tensor_load_to_lds(g0, g1, gz4, gz4, gz8, 0);
    }
    __builtin_amdgcn_s_wait_tensorcnt(0);
    __syncthreads();
#else
    for (int i = threadIdx.x; i < 4096; i += kBlock) s_l0[i] = emb[i];
    __syncthreads();
#endif

    int b = blockIdx.x * kBlock + threadIdx.x;
    if (b >= B) return;

    // x = (v + 1)/2 ; non-temporal: inputs are read once, keep L2 for the table
    const float* ip = in + 3u * (uint32_t)b;
    float x = fmaf(__builtin_nontemporal_load(ip + 0), 0.5f, 0.5f);
    float y = fmaf(__builtin_nontemporal_load(ip + 1), 0.5f, 0.5f);
    float z = fmaf(__builtin_nontemporal_load(ip + 2), 0.5f, 0.5f);

    float res[2 * kL];

#pragma unroll
    for (int l = 0; l < kL; ++l) {
        const float scale = (float)(16u << l) - 1.0f;
        float px = fmaf(x, scale, 0.5f);
        float py = fmaf(y, scale, 0.5f);
        float pz = fmaf(z, scale, 0.5f);
        float bxf = floorf(px), byf = floorf(py), bzf = floorf(pz);
        float fx = px - bxf, fy = py - byf, fz = pz - bzf;
        uint32_t ix = (uint32_t)bxf, iy = (uint32_t)byf, iz = (uint32_t)bzf;

        // fast_hash with PRIMES = (1, P1, P2); (c+1)*P == c*P + P (mod 2^32)
        uint32_t hy0 = iy * kP1, hy1 = hy0 + kP1;
        uint32_t hz0 = iz * kP2, hz1 = hz0 + kP2;
        uint32_t ix1 = ix + 1u;
        const uint32_t m = kMask[l];

        uint32_t i000 = (ix  ^ hy0 ^ hz0) & m;
        uint32_t i100 = (ix1 ^ hy0 ^ hz0) & m;
        uint32_t i010 = (ix  ^ hy1 ^ hz0) & m;
        uint32_t i110 = (ix1 ^ hy1 ^ hz0) & m;
        uint32_t i001 = (ix  ^ hy0 ^ hz1) & m;
        uint32_t i101 = (ix1 ^ hy0 ^ hz1) & m;
        uint32_t i011 = (ix  ^ hy1 ^ hz1) & m;
        uint32_t i111 = (ix1 ^ hy1 ^ hz1) & m;

        vf2 e000, e100, e010, e110, e001, e101, e011, e111;
        if (l == 0) {   // LDS-resident level
            e000 = s_l0[i000]; e100 = s_l0[i100];
            e010 = s_l0[i010]; e110 = s_l0[i110];
            e001 = s_l0[i001]; e101 = s_l0[i101];
            e011 = s_l0[i011]; e111 = s_l0[i111];
        } else {        // L2-resident gathers
            const vf2* __restrict__ p = emb + kOffset[l];
            e000 = p[i000]; e100 = p[i100];
            e010 = p[i010]; e110 = p[i110];
            e001 = p[i001]; e101 = p[i101];
            e011 = p[i011]; e111 = p[i111];
        }

        float gx = 1.0f - fx, gy = 1.0f - fy, gz = 1.0f - fz;
        float wy0z0 = gy * gz, wy1z0 = fy * gz;
        float wy0z1 = gy * fz, wy1z1 = fy * fz;
        float w000 = gx * wy0z0, w100 = fx * wy0z0;
        float w010 = gx * wy1z0, w110 = fx * wy1z0;
        float w001 = gx * wy0z1, w101 = fx * wy0z1;
        float w011 = gx * wy1z1, w111 = fx * wy1z1;

        float o0 = w000 * e000.x, o1 = w000 * e000.y;
        o0 = fmaf(w100, e100.x, o0); o1 = fmaf(w100, e100.y, o1);
        o0 = fmaf(w010, e010.x, o0); o1 = fmaf(w010, e010.y, o1);
        o0 = fmaf(w110, e110.x, o0); o1 = fmaf(w110, e110.y, o1);
        o0 = fmaf(w001, e001.x, o0); o1 = fmaf(w001, e001.y, o1);
        o0 = fmaf(w101, e101.x, o0); o1 = fmaf(w101, e101.y, o1);
        o0 = fmaf(w011, e011.x, o0); o1 = fmaf(w011, e011.y, o1);
        o0 = fmaf(w111, e111.x, o0); o1 = fmaf(w111, e111.y, o1);

        res[2 * l + 0] = o0;
        res[2 * l + 1] = o1;
    }

    // 128 B contiguous per point; NT stores (streamed once, keep L2 for table)
    float* op = out + (size_t)b * (2u * kL);
#pragma unroll
    for (int i = 0; i < 8; ++i) {
        vf4 v = {res[4 * i + 0], res[4 * i + 1], res[4 * i + 2], res[4 * i + 3]};
        __builtin_nontemporal_store(v, (vf4*)(op + 4 * i));
    }
}

extern "C" void kernel_launch(void* const* d_in, const int* in_sizes, int n_in,
                              void* d_out, int out_size, void* d_ws, size_t ws_size,
                              hipStream_t stream) {
    const float* inputs     = (const float*)d_in[0];   // [B, 3] float32
    const vf2*   embeddings = (const vf2*)d_in[1];     // [total_params, 2] float32
    float*       out        = (float*)d_out;           // [B, 32] float32
    int B = in_sizes[0] / 3;
    int grid = (B + kBlock - 1) / kBlock;
    grid_encode_kernel<<<grid, kBlock, 0, stream>>>(inputs, embeddings, out, B);
}